// ESN_19585050870243
// MI455X (gfx1250) — compile-verified
//
#include <hip/hip_runtime.h>
#include <hip/hip_bf16.h>

// ---------------- problem constants ----------------
#define RES        2048
#define INSZ       128
#define TILES_N    (RES / 16)          // 128
#define LEAK       0.9f
#define ONE_MINUS_LEAK 0.1f
#define INV_SQRT_N 0.02209708691207961f   // 1/sqrt(2048)

// ---------------- recurrence config ----------------
#define NWG         64                 // persistent workgroups
#define TPB         256                // 8 waves (wave32)
#define ROWS_PER_WG (RES / NWG)        // 32
#define LPR         8                  // lanes cooperating on one row
#define NITER       (RES / (LPR * 4))  // 64 float4 iterations per lane

typedef __attribute__((ext_vector_type(2))) float v2f;
typedef __attribute__((ext_vector_type(8))) float v8f;

// ---------------------------------------------------------------------------
// Kernel 0: zero the state double-buffer and barrier words (d_ws is poisoned).
// ---------------------------------------------------------------------------
__global__ void esn_init_kernel(float* __restrict__ gx, int* __restrict__ bar) {
    int tid = threadIdx.x;
    #pragma unroll
    for (int i = 0; i < (2 * RES) / TPB; ++i)
        gx[tid + i * TPB] = 0.0f;
    if (tid < 2) bar[tid] = 0;
}

// ---------------------------------------------------------------------------
// Kernel 1: proj = input @ W_in^T  via V_WMMA_F32_16X16X4_F32 (full f32).
// One wave computes one 16x16 tile of proj; K=128 -> 32 chained WMMAs.
//   A (16x4 f32): lane l<16 holds row M=l, K={0,1}; lane>=16 holds K={2,3}.
//   B (4x16 f32): lane l<16 holds col N=l, K={0,1}; lane>=16 holds K={2,3}.
//   C (16x16 f32): VGPR v: lanes 0-15 -> M=v, lanes 16-31 -> M=v+8, N=lane%16.
// Writes proj directly into d_out[t, r] (recurrence transforms it in place).
// ---------------------------------------------------------------------------
__global__ void esn_proj_wmma_kernel(const float* __restrict__ A,      // [T, 128]
                                     const float* __restrict__ Win,    // [R, 128]
                                     float* __restrict__ out,          // [T, R]
                                     int T) {
    const int lane  = threadIdx.x & 31;
    const int wave  = threadIdx.x >> 5;
    const int gw    = blockIdx.x * 8 + wave;     // global wave id = tile id
    const int tM    = gw / TILES_N;
    const int tN    = gw % TILES_N;
    if (tM * 16 >= T) return;                    // uniform per wave

    const int lhalf = lane >> 4;                 // 0 or 1 (K-pair select)
    const int l16   = lane & 15;

    const float* aptr = A   + (size_t)(tM * 16 + l16) * INSZ + lhalf * 2;
    const float* bptr = Win + (size_t)(tN * 16 + l16) * INSZ + lhalf * 2;

    v8f c = {};
    #pragma unroll
    for (int k = 0; k < INSZ; k += 4) {
        v2f a, b;
        a.x = aptr[k];  a.y = aptr[k + 1];
        b.x = bptr[k];  b.y = bptr[k + 1];
        c = __builtin_amdgcn_wmma_f32_16x16x4_f32(
                /*neg_a=*/false, a, /*neg_b=*/false, b,
                /*c_mod=*/(short)0, c, /*reuse_a=*/false, /*reuse_b=*/false);
    }

    float* obase = out + (size_t)(tM * 16) * RES + tN * 16 + l16;
    #pragma unroll
    for (int v = 0; v < 8; ++v) {
        int m = v + lhalf * 8;
        obase[(size_t)m * RES] = c[v];
    }
}

// ---------------------------------------------------------------------------
// Kernel 2: persistent sequential scan.
//   64 WGs own 32 rows of W_res each (256 KB/step streamed from L2-resident
//   W_res). Per step: stage x_prev (8 KB) into LDS, 8-lane float4 dot per row,
//   shfl_xor reduce, erf + leaky update, write x_t over proj row in d_out and
//   into the double-buffered global state, then agent-scope split barrier.
// ---------------------------------------------------------------------------
__global__ void esn_recur_kernel(const float* __restrict__ Wres,  // [R, R]
                                 float* __restrict__ out,         // [T, R] proj->x
                                 float* __restrict__ gx,          // [2][R]
                                 int* __restrict__ bar_cnt,
                                 int* __restrict__ bar_gen,
                                 int T) {
    __shared__ float xprev[RES];

    const int tid    = threadIdx.x;
    const int wg     = blockIdx.x;
    const int sub    = tid & (LPR - 1);     // 0..7: lane within row group
    const int rlocal = tid >> 3;            // 0..31: row within slice
    const int row    = wg * ROWS_PER_WG + rlocal;

    const float4* wrow = (const float4*)(Wres + (size_t)row * RES);

    for (int t = 0; t < T; ++t) {
        const float4* xsrc = (const float4*)(gx + (size_t)(t & 1) * RES);
        float*        xdst = gx + (size_t)((t + 1) & 1) * RES;

        // stage x_prev into LDS (2048 floats, 2 x float4 per thread)
        float4* l4 = (float4*)xprev;
        l4[tid]       = xsrc[tid];
        l4[tid + TPB] = xsrc[tid + TPB];
        __syncthreads();

        // slice mat-vec: 8 lanes x 64 float4 iterations cover one 2048-col row
        float acc = 0.0f;
        const float4* x4 = (const float4*)xprev;
        #pragma unroll 8
        for (int i = 0; i < NITER; ++i) {
            float4 w  = wrow[i * LPR + sub];
            float4 xv = x4[i * LPR + sub];
            acc = fmaf(w.x, xv.x, acc);
            acc = fmaf(w.y, xv.y, acc);
            acc = fmaf(w.z, xv.z, acc);
            acc = fmaf(w.w, xv.w, acc);
        }
        // reduce the 8-lane group (within wave32)
        acc += __shfl_xor(acc, 1, 32);
        acc += __shfl_xor(acc, 2, 32);
        acc += __shfl_xor(acc, 4, 32);

        if (sub == 0) {
            size_t oi  = (size_t)t * RES + row;
            float pre  = out[oi] + acc;                 // proj[t,row] + W_res@x
            float xp   = xprev[row];
            float xn   = ONE_MINUS_LEAK * xp + LEAK * erff(pre) * INV_SQRT_N;
            out[oi]    = xn;
            xdst[row]  = xn;
        }

        // device-wide split barrier (arrive + spin on generation)
        __syncthreads();
        if (tid == 0) {
            __threadfence();
            int g = __hip_atomic_load(bar_gen, __ATOMIC_ACQUIRE,
                                      __HIP_MEMORY_SCOPE_AGENT);
            int arrived = __hip_atomic_fetch_add(bar_cnt, 1, __ATOMIC_ACQ_REL,
                                                 __HIP_MEMORY_SCOPE_AGENT);
            if (arrived == NWG - 1) {
                __hip_atomic_store(bar_cnt, 0, __ATOMIC_RELAXED,
                                   __HIP_MEMORY_SCOPE_AGENT);
                __hip_atomic_fetch_add(bar_gen, 1, __ATOMIC_RELEASE,
                                       __HIP_MEMORY_SCOPE_AGENT);
            } else {
                while (__hip_atomic_load(bar_gen, __ATOMIC_ACQUIRE,
                                         __HIP_MEMORY_SCOPE_AGENT) == g) {
                    __builtin_amdgcn_s_sleep(2);
                }
            }
        }
        __syncthreads();
    }
}

// ---------------------------------------------------------------------------
extern "C" void kernel_launch(void* const* d_in, const int* in_sizes, int n_in,
                              void* d_out, int out_size, void* d_ws, size_t ws_size,
                              hipStream_t stream) {
    const float* input = (const float*)d_in[0];   // [T, 128]
    const float* Win   = (const float*)d_in[1];   // [2048, 128]
    const float* Wres  = (const float*)d_in[2];   // [2048, 2048]
    float*       out   = (float*)d_out;           // [T, 2048]

    const int T = in_sizes[0] / INSZ;             // 20000

    // d_ws layout: [2*RES floats: x double-buffer][2 ints: barrier cnt/gen]
    float* gx  = (float*)d_ws;
    int*   bar = (int*)((char*)d_ws + (size_t)2 * RES * sizeof(float));

    esn_init_kernel<<<dim3(1), dim3(TPB), 0, stream>>>(gx, bar);

    const int nTiles  = (T / 16) * TILES_N;       // 1250 * 128 = 160000
    const int nBlocks = (nTiles + 7) / 8;         // 8 waves (tiles) per block
    esn_proj_wmma_kernel<<<dim3(nBlocks), dim3(TPB), 0, stream>>>(input, Win, out, T);

    esn_recur_kernel<<<dim3(NWG), dim3(TPB), 0, stream>>>(Wres, out, gx,
                                                          bar, bar + 1, T);
}